// GraphBaseClass_v1_50568944943199
// MI455X (gfx1250) — compile-verified
//
#include <hip/hip_runtime.h>
#include <hip/hip_bf16.h>

// MI455X / gfx1250, wave32.
//
// out[i_local, j] = dis_sq if valid else 0, shape (chunk=2048, n=16384) f32.
// dis_sq computed as (n_j - 2 p_i.p_j) + n_i via V_WMMA_F32_16X16X4_F32:
//   A (16x4):  row i = (-2x_i, -2y_i, -2z_i, 1)
//   B (4x16):  col j = ( x_j,   y_j,   z_j, n_j)
// Hash-neighbor mask replaced by exact voxel adjacency (mathematically
// equivalent modulo 2^-32-probability 32-bit hash collisions; the hash is
// linear in voxel coords so adjacency <=> hash delta in the 27-value set).

typedef __attribute__((ext_vector_type(2))) float v2f;
typedef __attribute__((ext_vector_type(8))) float v8f;

#define COLS_PER_BLOCK 512   // 8 waves * 4 j-tiles * 16 cols
#define JTILES_PER_WAVE 4

__global__ __launch_bounds__(256)
void edge_dist_wmma_kernel(const float* __restrict__ points,
                           const float* __restrict__ radii,
                           const int* __restrict__ startp,
                           float* __restrict__ out,
                           int n, int chunk)
{
    // i-tile staging: A coefficients + norm, voxel ints + global index, radius
    __shared__ float sA[16][4];   // -2x, -2y, -2z, |p|^2
    __shared__ int   sV[16][4];   // vx, vy, vz, global idx
    __shared__ float sR[16];      // radius

    const int start  = *startp;
    const int itile  = blockIdx.y;                 // 16 output rows
    const int tid    = threadIdx.x;
    const int lane   = tid & 31;
    const int wave   = tid >> 5;
    const int col    = lane & 15;                  // N within tile / M for A
    const int half   = lane >> 4;                  // 0: lanes 0-15, 1: 16-31

    if (tid < 16) {
        int iLocal = itile * 16 + tid;
        int gi = start + iLocal;
        float x = points[gi * 3 + 0];
        float y = points[gi * 3 + 1];
        float z = points[gi * 3 + 2];
        sA[tid][0] = -2.0f * x;
        sA[tid][1] = -2.0f * y;
        sA[tid][2] = -2.0f * z;
        sA[tid][3] = x * x + y * y + z * z;
        sV[tid][0] = (int)x;       // VOXEL_SIZE == 1.0, trunc toward zero
        sV[tid][1] = (int)y;
        sV[tid][2] = (int)z;
        sV[tid][3] = gi;
        sR[tid] = radii[gi];
    }
    __syncthreads();

    // A operand (constant across the whole j-loop).
    // f32 16x4 A layout: lanes 0-15 -> M=lane, VGPR0=K0, VGPR1=K1;
    //                    lanes 16-31 -> M=lane-16, VGPR0=K2, VGPR1=K3.
    v2f a;
    a.x = half ? sA[col][2] : sA[col][0];
    a.y = half ? 1.0f       : sA[col][1];

    // Hoist per-row epilogue constants into VGPRs (rows rr + 8*half).
    float ni[8], ri_[8];
    int   vxi[8], vyi[8], vzi[8], gii[8];
#pragma unroll
    for (int rr = 0; rr < 8; ++rr) {
        int r   = rr + 8 * half;
        ni[rr]  = sA[r][3];
        vxi[rr] = sV[r][0];
        vyi[rr] = sV[r][1];
        vzi[rr] = sV[r][2];
        gii[rr] = sV[r][3];
        ri_[rr] = sR[r];
    }

    const int jbase0 = blockIdx.x * COLS_PER_BLOCK + wave * (JTILES_PER_WAVE * 16);

#pragma unroll
    for (int t = 0; t < JTILES_PER_WAVE; ++t) {
        const int jbase = jbase0 + t * 16;
        const int j = jbase + col;               // this lane's column

        // j-point attributes (both lane halves fetch the same 16 columns; L2 hit)
        float xj = points[j * 3 + 0];
        float yj = points[j * 3 + 1];
        float zj = points[j * 3 + 2];
        float nj = xj * xj + yj * yj + zj * zj;
        float rj = radii[j];
        int vxj = (int)xj, vyj = (int)yj, vzj = (int)zj;

        // B operand, mirror of A layout: lanes 0-15 -> K0/K1, lanes 16-31 -> K2/K3.
        v2f b;
        b.x = half ? zj : xj;
        b.y = half ? nj : yj;

        v8f acc = {};
        // D = A x B : acc[rr] = n_j - 2 p_i.p_j   for row (rr + 8*half), col j
        acc = __builtin_amdgcn_wmma_f32_16x16x4_f32(
            /*neg_a=*/false, a, /*neg_b=*/false, b,
            /*c_mod=*/(short)0, acc, /*reuse_a=*/false, /*reuse_b=*/false);

#pragma unroll
        for (int rr = 0; rr < 8; ++rr) {
            float dis = acc[rr] + ni[rr];

            // voxel adjacency (== 27-way hash match modulo collisions)
            int dx = vxj - vxi[rr];
            int dy = vyj - vyi[rr];
            int dz = vzj - vzi[rr];
            int adj = ((unsigned)(dx + 1) <= 2u) &
                      ((unsigned)(dy + 1) <= 2u) &
                      ((unsigned)(dz + 1) <= 2u);

            // radius term: max_r = 1.5*min(ri,rj); s = min(ri,max_r)+min(rj,max_r)
            float ri   = ri_[rr];
            float rmin = fminf(ri, rj);
            float maxr = rmin * 1.5f;
            float s    = fminf(ri, maxr) + fminf(rj, maxr);

            int valid = adj & (j > gii[rr]) & (dis < s * s) & (ri < 1.0f);

            int iLocal = itile * 16 + rr + 8 * half;
            out[(size_t)iLocal * (size_t)n + (size_t)j] = valid ? dis : 0.0f;
        }
    }
}

extern "C" void kernel_launch(void* const* d_in, const int* in_sizes, int n_in,
                              void* d_out, int out_size, void* d_ws, size_t ws_size,
                              hipStream_t stream) {
    const float* points = (const float*)d_in[0];   // (n, 3) f32
    const float* radii  = (const float*)d_in[1];   // (n, 1) f32
    const int*   startp = (const int*)d_in[2];     // scalar int (device)
    // d_in[3] = end_idx (unused on host; chunk derived from out_size)

    const int n     = in_sizes[0] / 3;             // 16384
    const int chunk = out_size / n;                // 2048

    dim3 block(256);                               // 8 wave32 waves
    dim3 grid((unsigned)(n / COLS_PER_BLOCK),      // 32 column blocks
              (unsigned)(chunk / 16));             // 128 i-tiles

    edge_dist_wmma_kernel<<<grid, block, 0, stream>>>(
        points, radii, startp, (float*)d_out, n, chunk);
}